// QMLP_14654428414265
// MI455X (gfx1250) — compile-verified
//
#include <hip/hip_runtime.h>
#include <hip/hip_bf16.h>
#include <math.h>

#define N_QUBITS 9
#define N_LAYERS 5
#define NUM_CLASSES 10
#define BATCH 4096
#define EV_STRIDE 12

typedef __attribute__((ext_vector_type(2))) float v2f;
typedef __attribute__((ext_vector_type(8))) float v8f;

struct C2 { float x, y; };
__device__ __forceinline__ C2 cmul(C2 a, C2 b) {
  return { a.x * b.x - a.y * b.y, a.x * b.y + a.y * b.x };
}

// ---------------------------------------------------------------------------
// State layout (per wave = one batch sample):
//   amplitude index idx = lane*16 + r   (lane = bits 8..4, r = bits 3..0)
//   wire w (PennyLane: wire 0 most significant) lives on bit B = 8 - w.
//   B < 4  -> register-pair gate (pure VALU)
//   B >= 4 -> cross-lane gate via __shfl_xor (wave32 butterfly)
// ---------------------------------------------------------------------------

// Generic single-qubit gate (used for Rot)
template<int B>
__device__ __forceinline__ void apply1(float* sr, float* si,
                                       C2 g00, C2 g01, C2 g10, C2 g11, int lane) {
  if constexpr (B < 4) {
    constexpr int m = 1 << B;
#pragma unroll
    for (int r0 = 0; r0 < 16; ++r0) {
      if ((r0 & m) == 0) {
        const int r1 = r0 | m;
        C2 a0{ sr[r0], si[r0] }, a1{ sr[r1], si[r1] };
        C2 t00 = cmul(g00, a0), t01 = cmul(g01, a1);
        C2 t10 = cmul(g10, a0), t11 = cmul(g11, a1);
        sr[r0] = t00.x + t01.x; si[r0] = t00.y + t01.y;
        sr[r1] = t10.x + t11.x; si[r1] = t10.y + t11.y;
      }
    }
  } else {
    constexpr int lm = 1 << (B - 4);
    const bool hi = (lane & lm) != 0;
    C2 cme = hi ? g11 : g00;
    C2 cpa = hi ? g10 : g01;
#pragma unroll
    for (int r = 0; r < 16; ++r) {
      float pr = __shfl_xor(sr[r], lm, 32);
      float pi = __shfl_xor(si[r], lm, 32);
      C2 me{ sr[r], si[r] }, pa{ pr, pi };
      C2 t0 = cmul(cme, me), t1 = cmul(cpa, pa);
      sr[r] = t0.x + t1.x; si[r] = t0.y + t1.y;
    }
  }
}

// RX(theta): g00=g11=(c,0), g01=g10=(0,-s) — symmetric, so lane path needs no hi/lo split
template<int B>
__device__ __forceinline__ void rxgate(float* sr, float* si, float c, float s, int lane) {
  if constexpr (B < 4) {
    constexpr int m = 1 << B;
#pragma unroll
    for (int r0 = 0; r0 < 16; ++r0) {
      if ((r0 & m) == 0) {
        const int r1 = r0 | m;
        float a0r = sr[r0], a0i = si[r0], a1r = sr[r1], a1i = si[r1];
        sr[r0] = c * a0r + s * a1i;  si[r0] = c * a0i - s * a1r;
        sr[r1] = c * a1r + s * a0i;  si[r1] = c * a1i - s * a0r;
      }
    }
  } else {
    constexpr int lm = 1 << (B - 4);
#pragma unroll
    for (int r = 0; r < 16; ++r) {
      float pr = __shfl_xor(sr[r], lm, 32);
      float pi = __shfl_xor(si[r], lm, 32);
      float nr = c * sr[r] + s * pi;
      float ni = c * si[r] - s * pr;
      sr[r] = nr; si[r] = ni;
    }
  }
}

// CRX: apply RX on target bit BT only where control bit BC == 1
template<int BC, int BT>
__device__ __forceinline__ void crxgate(float* sr, float* si, float c, float s, int lane) {
  if constexpr (BT < 4) {
    constexpr int m = 1 << BT;
    float cc = c, ss = s;
    if constexpr (BC >= 4) {
      const bool act = (lane & (1 << (BC - 4))) != 0;
      cc = act ? c : 1.0f;
      ss = act ? s : 0.0f;
    }
#pragma unroll
    for (int r0 = 0; r0 < 16; ++r0) {
      if ((r0 & m) == 0) {
        if constexpr (BC < 4) { if ((r0 & (1 << BC)) == 0) continue; }
        const int r1 = r0 | m;
        float a0r = sr[r0], a0i = si[r0], a1r = sr[r1], a1i = si[r1];
        sr[r0] = cc * a0r + ss * a1i;  si[r0] = cc * a0i - ss * a1r;
        sr[r1] = cc * a1r + ss * a0i;  si[r1] = cc * a1i - ss * a0r;
      }
    }
  } else {
    constexpr int lm = 1 << (BT - 4);
#pragma unroll
    for (int r = 0; r < 16; ++r) {
      float pr = __shfl_xor(sr[r], lm, 32);
      float pi = __shfl_xor(si[r], lm, 32);
      bool act;
      if constexpr (BC < 4) act = (r & (1 << BC)) != 0;
      else                  act = (lane & (1 << (BC - 4))) != 0;
      float nr = c * sr[r] + s * pi;
      float ni = c * si[r] - s * pr;
      sr[r] = act ? nr : sr[r];
      si[r] = act ? ni : si[r];
    }
  }
}

// ---------------------------------------------------------------------------
// Kernel 1: full 9-qubit circuit, one wave per sample, state in 32 VGPRs.
// Writes <Z_w> to evs[sample*12 + w], cols 9..11 zeroed (K-padding for WMMA).
// ---------------------------------------------------------------------------
__global__ __launch_bounds__(256) void QMLP_sim(const float* __restrict__ x,
                                                const float* __restrict__ rot_w,
                                                const float* __restrict__ crx_w,
                                                float* __restrict__ evs) {
  const int lane = threadIdx.x & 31;
  const int sample = blockIdx.x * 8 + (threadIdx.x >> 5);

  float sr[16], si[16];
#pragma unroll
  for (int r = 0; r < 16; ++r) { sr[r] = 0.0f; si[r] = 0.0f; }
  if (lane == 0) sr[0] = 1.0f;   // |0...0>

  // AngleEmbedding angles reused each layer: precompute cos/sin once
  float cx[9], sx[9];
#pragma unroll
  for (int w = 0; w < 9; ++w) {
    float th = 0.5f * x[sample * 9 + w];
    sincosf(th, &sx[w], &cx[w]);
  }

  for (int n = 0; n < N_LAYERS; ++n) {
    const float* rw = rot_w + n * 27;
    const float* cw = crx_w + n * 9;

    // ---- AngleEmbedding RX on all wires ----
#define RX_G(W) rxgate<8 - (W)>(sr, si, cx[W], sx[W], lane);
    RX_G(0) RX_G(1) RX_G(2) RX_G(3) RX_G(4) RX_G(5) RX_G(6) RX_G(7) RX_G(8)
#undef RX_G

    // ---- Rot = RZ(omega) RY(theta) RZ(phi) ----
#define ROT_G(W) { \
      float phi = rw[(W) * 3 + 0], th = rw[(W) * 3 + 1], om = rw[(W) * 3 + 2]; \
      float st, ct; sincosf(0.5f * th, &st, &ct); \
      float sp, cp; sincosf(0.5f * (phi + om), &sp, &cp); \
      float sm, cm; sincosf(0.5f * (phi - om), &sm, &cm); \
      C2 g00{ ct * cp, -ct * sp };  C2 g01{ -st * cm, -st * sm }; \
      C2 g10{ st * cm, -st * sm };  C2 g11{ ct * cp,  ct * sp }; \
      apply1<8 - (W)>(sr, si, g00, g01, g10, g11, lane); }
    ROT_G(0) ROT_G(1) ROT_G(2) ROT_G(3) ROT_G(4) ROT_G(5) ROT_G(6) ROT_G(7) ROT_G(8)
#undef ROT_G

    // ---- CRX ring entanglers: ctrl w, tgt (w+1)%9 ----
#define CRX_G(W) { \
      float th = 0.5f * cw[W]; float st, ct; sincosf(th, &st, &ct); \
      crxgate<8 - (W), 8 - (((W) + 1) % 9)>(sr, si, ct, st, lane); }
    CRX_G(0) CRX_G(1) CRX_G(2) CRX_G(3) CRX_G(4) CRX_G(5) CRX_G(6) CRX_G(7) CRX_G(8)
#undef CRX_G
  }

  // ---- probabilities and <Z_w> ----
  float pv[16];
  float ptot = 0.0f;
#pragma unroll
  for (int r = 0; r < 16; ++r) { pv[r] = sr[r] * sr[r] + si[r] * si[r]; ptot += pv[r]; }

  float ev[9];
  // wires 0..4: bit 8-w is a lane bit (lane bit 4-w); sign uniform over r
#pragma unroll
  for (int w = 0; w < 5; ++w) {
    const int lb = 4 - w;
    ev[w] = ((lane >> lb) & 1) ? -ptot : ptot;
  }
  // wires 5..8: bit 8-w is a register bit (3..0)
#pragma unroll
  for (int w = 5; w < 9; ++w) {
    const int b = 8 - w;
    float s = 0.0f;
#pragma unroll
    for (int r = 0; r < 16; ++r) s += ((r >> b) & 1) ? -pv[r] : pv[r];
    ev[w] = s;
  }

  // wave-wide butterfly reduction over 32 lanes (all lanes end with full sums)
#pragma unroll
  for (int mask = 16; mask >= 1; mask >>= 1) {
#pragma unroll
    for (int w = 0; w < 9; ++w) ev[w] += __shfl_xor(ev[w], mask, 32);
  }

  if (lane == 0) {
#pragma unroll
    for (int w = 0; w < 9; ++w) evs[sample * EV_STRIDE + w] = ev[w];
    evs[sample * EV_STRIDE + 9]  = 0.0f;
    evs[sample * EV_STRIDE + 10] = 0.0f;
    evs[sample * EV_STRIDE + 11] = 0.0f;
  }
}

// ---------------------------------------------------------------------------
// Kernel 2: logits = evs @ fc_w^T + fc_b via V_WMMA_F32_16X16X4_F32
// (M=16 batch rows/tile, N=16 (10 classes + pad), K=12 = 3 x k-tile of 4),
// then per-row log_softmax through LDS.
// A frag: lane L holds A[M=L&15, K=2*(L>>4)+v] ; B frag: B[K=2*(L>>4)+v, N=L&15]
// D tile: VGPR v holds D[M=v+8*(L>>4), N=L&15]
// ---------------------------------------------------------------------------
__global__ __launch_bounds__(128) void QMLP_fc(const float* __restrict__ evs,
                                               const float* __restrict__ fc_w,
                                               const float* __restrict__ fc_b,
                                               float* __restrict__ out) {
  __shared__ float tile[4][16][17];
  const int lane = threadIdx.x & 31;
  const int wave = threadIdx.x >> 5;
  const int t    = blockIdx.x * 4 + wave;
  const int row0 = t * 16;
  const int mn   = lane & 15;   // M for A-frag, N for B/C/D
  const int half = lane >> 4;

  v8f c;
  const float bias = (mn < NUM_CLASSES) ? fc_b[mn] : 0.0f;
#pragma unroll
  for (int v = 0; v < 8; ++v) c[v] = bias;

#pragma unroll
  for (int kt = 0; kt < 3; ++kt) {
    v2f a, b;
#pragma unroll
    for (int v = 0; v < 2; ++v) {
      const int k = kt * 4 + half * 2 + v;
      a[v] = evs[(row0 + mn) * EV_STRIDE + k];                 // cols 9..11 are zero
      b[v] = (mn < NUM_CLASSES && k < N_QUBITS) ? fc_w[mn * N_QUBITS + k] : 0.0f;
    }
    // D = A x B + C, f32 accumulate
    c = __builtin_amdgcn_wmma_f32_16x16x4_f32(false, a, false, b, (short)0, c,
                                              false, false);
  }

  // scatter D tile to LDS so each row sits contiguously for log_softmax
#pragma unroll
  for (int v = 0; v < 8; ++v) tile[wave][v + 8 * half][mn] = c[v];
  __syncthreads();

  if (lane < 16) {
    const int row = lane;
    float vals[NUM_CLASSES];
    float m = -3.0e38f;
#pragma unroll
    for (int j = 0; j < NUM_CLASSES; ++j) {
      vals[j] = tile[wave][row][j];
      m = fmaxf(m, vals[j]);
    }
    float ssum = 0.0f;
#pragma unroll
    for (int j = 0; j < NUM_CLASSES; ++j) ssum += expf(vals[j] - m);
    const float lse = m + logf(ssum);
#pragma unroll
    for (int j = 0; j < NUM_CLASSES; ++j)
      out[(row0 + row) * NUM_CLASSES + j] = vals[j] - lse;
  }
}

extern "C" void kernel_launch(void* const* d_in, const int* in_sizes, int n_in,
                              void* d_out, int out_size, void* d_ws, size_t ws_size,
                              hipStream_t stream) {
  const float* x     = (const float*)d_in[0];   // [4096, 9]
  const float* rot_w = (const float*)d_in[1];   // [5, 9, 3]
  const float* crx_w = (const float*)d_in[2];   // [5, 9, 1]
  const float* fc_w  = (const float*)d_in[3];   // [10, 9]
  const float* fc_b  = (const float*)d_in[4];   // [10]
  float* out = (float*)d_out;                   // [4096, 10]
  float* evs = (float*)d_ws;                    // [4096, 12] padded <Z> scratch

  QMLP_sim<<<BATCH / 8, 256, 0, stream>>>(x, rot_w, crx_w, evs);
  QMLP_fc<<<(BATCH / 16) / 4, 128, 0, stream>>>(evs, fc_w, fc_b, out);
}